// ModelNew_50388556316937
// MI455X (gfx1250) — compile-verified
//
#include <hip/hip_runtime.h>
#include <hip/hip_bf16.h>

typedef __attribute__((ext_vector_type(2))) float v2f;
typedef __attribute__((ext_vector_type(8))) float v8f;

// ---- geometry constants ----
// x: (16, 16, 16, 32, 32)  NCDHW
// w: (16, 32, 3, 3, 3)     (in, out, kd, kh, kw)
// y: (16, 32, 31, 63, 63)
#define XN 16
#define XC 16
#define XD 16
#define XH 32
#define XW 32
#define OC 32
#define OD 31
#define OH 63
#define OW 63
#define X_CSTRIDE (XD*XH*XW)      // 16384
#define Y_CSTRIDE (OD*OH*OW)      // 123039
#define W_CISTRIDE (OC*27)        // 864

// -----------------------------------------------------------------------------
// ConvTranspose3d via parity decomposition + V_WMMA_F32_16X16X4_F32.
// D/H parity is a template parameter; BOTH W parities are computed by the same
// wave so each (co,row) gets a fully dense 128B store from one wave.
//
// Output W coords: ow = 2*qw   (even, tap kw=1, iw=qw)
//                  ow = 2*qw+1 (odd,  taps kw=2 iw=qw  and  kw=0 iw=qw+1)
// A fragment layout (ISA 7.12.2, 32-bit A 16x4): lanes 0-15 hold M=lane,
// K = {0,1} in the two VGPRs; lanes 16-31 hold M=lane-16, K = {2,3}.
// B mirrors this with N across lanes 0-15.
//
// LDS weight layout: wlds[tap][ci/2][co][ci%2] so every B fragment (the K
// pair {ci0, ci0+1} at one co) is a single aligned float2 -> one ds_load_b64.
// -----------------------------------------------------------------------------
template<int PD, int PH>
__global__ __launch_bounds__(256) void convt_wmma_kernel(
    const float* __restrict__ x, const float* __restrict__ w,
    float* __restrict__ y)
{
  // 27 taps * 8 ci-pairs * 32 co * 2 = 13824 floats (54KB)
  __shared__ float wlds[27 * 16 * 32];
  const int tid = threadIdx.x;
  for (int i = tid; i < 27 * 512; i += 256) {
    const int tap = i >> 9;          // i = tap*512 + p*64 + co*2 + lo
    const int rem = i & 511;
    const int p   = rem >> 6;        // ci pair index
    const int q   = rem & 63;
    const int co  = q >> 1;
    const int lo  = q & 1;
    const int ci  = 2 * p + lo;
    wlds[i] = w[ci * W_CISTRIDE + co * 27 + tap];
  }
  __syncthreads();

  const int lane = tid & 31;
  const int wave = tid >> 5;
  const int m    = lane & 15;   // position / N index within half-wave
  const int half = lane >> 4;   // 0: K pair {0,1}, 1: K pair {2,3}

  constexpr int Dq = PD ? 15 : 16;   // # output coords of this parity per dim
  constexpr int Hq = PH ? 31 : 32;
  constexpr int TILES = XN * Dq * Hq * 2;   // 2 W-tiles of 16 qw each

  const int tile = blockIdx.x * 8 + wave;
  if (tile >= TILES) return;

  const int qwt = tile & 1;
  int t1 = tile >> 1;
  const int qh = t1 % Hq;  t1 /= Hq;
  const int qd = t1 % Dq;
  const int n  = t1 / Dq;
  const int qw0 = qwt * 16;

  v8f aE0 = {};   // even ow, co 0..15
  v8f aE1 = {};   // even ow, co 16..31
  v8f aO0 = {};   // odd  ow, co 0..15
  v8f aO1 = {};   // odd  ow, co 16..31

  const float* xn = x + (size_t)n * (XC * X_CSTRIDE);

  // Per-dim taps for D/H: parity 0 -> {k=1,d=0}; parity 1 -> {k=2,d=0},{k=0,d=1}
  constexpr int NTD = PD ? 2 : 1;
  constexpr int NTH = PH ? 2 : 1;
  const int kds[2] = { PD ? 2 : 1, 0 };
  const int khs[2] = { PH ? 2 : 1, 0 };

  // W-dim input coords for this lane
  const int iw0  = qw0 + m;            // delta_w = 0; always < 32
  const int iw1  = iw0 + 1;            // delta_w = 1
  const bool w1ok = (iw1 < XW);        // clamp+mask boundary lane
  const int iw1c = w1ok ? iw1 : (XW - 1);

#pragma unroll
  for (int td = 0; td < NTD; ++td) {
    const int id = qd + td;
    const int kd = kds[td];
#pragma unroll
    for (int th = 0; th < NTH; ++th) {
      const int ih = qh + th;
      const int kh = khs[th];
      const float* xs0 = xn + id * (XH * XW) + ih * XW + iw0;
      const float* xs1 = xn + id * (XH * XW) + ih * XW + iw1c;
      const v2f* wtE  = (const v2f*)(wlds + (kd * 9 + kh * 3 + 1) * 512);
      const v2f* wtO2 = (const v2f*)(wlds + (kd * 9 + kh * 3 + 2) * 512);
      const v2f* wtO0 = (const v2f*)(wlds + (kd * 9 + kh * 3 + 0) * 512);
#pragma unroll
      for (int s = 0; s < 4; ++s) {
        const int ci0 = 4 * s + 2 * half;    // this lane's K pair base
        const int p32 = (2 * s + half) * 32; // float2 row of this K pair
        v2f a0, a1;
        a0.x = xs0[(ci0    ) * X_CSTRIDE];
        a0.y = xs0[(ci0 + 1) * X_CSTRIDE];
        const float t0  = xs1[(ci0    ) * X_CSTRIDE];
        const float t1v = xs1[(ci0 + 1) * X_CSTRIDE];
        a1.x = w1ok ? t0  : 0.0f;
        a1.y = w1ok ? t1v : 0.0f;

        const v2f bE0 = wtE [p32 + m];
        const v2f bE1 = wtE [p32 + 16 + m];
        const v2f b20 = wtO2[p32 + m];
        const v2f b21 = wtO2[p32 + 16 + m];
        const v2f b00 = wtO0[p32 + m];
        const v2f b01 = wtO0[p32 + 16 + m];

        aE0 = __builtin_amdgcn_wmma_f32_16x16x4_f32(
            false, a0, false, bE0, (short)0, aE0, false, false);
        aE1 = __builtin_amdgcn_wmma_f32_16x16x4_f32(
            false, a0, false, bE1, (short)0, aE1, false, false);
        aO0 = __builtin_amdgcn_wmma_f32_16x16x4_f32(
            false, a0, false, b20, (short)0, aO0, false, false);
        aO1 = __builtin_amdgcn_wmma_f32_16x16x4_f32(
            false, a0, false, b21, (short)0, aO1, false, false);
        aO0 = __builtin_amdgcn_wmma_f32_16x16x4_f32(
            false, a1, false, b00, (short)0, aO0, false, false);
        aO1 = __builtin_amdgcn_wmma_f32_16x16x4_f32(
            false, a1, false, b01, (short)0, aO1, false, false);
      }
    }
  }

  // Store: C/D layout — lane holds N = lane&15 (co), VGPR v -> M = v + 8*half.
  // Even+odd interleave => wave writes 32 consecutive dwords per co-row.
  const int od = 2 * qd + PD;
  const int oh = 2 * qh + PH;
  const size_t obase = (size_t)n * (OC * (size_t)Y_CSTRIDE)
                     + (size_t)od * (OH * OW) + (size_t)oh * OW;
  const size_t c0off = (size_t)(m)      * Y_CSTRIDE;
  const size_t c1off = (size_t)(16 + m) * Y_CSTRIDE;
#pragma unroll
  for (int v = 0; v < 8; ++v) {
    const int mrow = v + 8 * half;
    const int qw = qw0 + mrow;
    const int ow = 2 * qw;               // 0..62, always valid
    y[obase + c0off + ow] = aE0[v];
    y[obase + c1off + ow] = aE1[v];
    if (qw < 31) {                        // ow+1 <= 62
      y[obase + c0off + ow + 1] = aO0[v];
      y[obase + c1off + ow + 1] = aO1[v];
    }
  }
}

// -----------------------------------------------------------------------------
// BN stage 1: per (channel, n) block reduces one contiguous 123039-float slab
// to (sum, sumsq) in f64. Fixed tree order => deterministic.
// -----------------------------------------------------------------------------
__global__ __launch_bounds__(256) void bn_stats_kernel(
    const float* __restrict__ y, double* __restrict__ part)
{
  const int c = blockIdx.x;   // 0..31
  const int n = blockIdx.y;   // 0..15
  const float* p = y + ((size_t)n * OC + c) * Y_CSTRIDE;
  double s = 0.0, sq = 0.0;
  for (int i = threadIdx.x; i < Y_CSTRIDE; i += 256) {
    const double v = (double)p[i];
    s += v;
    sq += v * v;
  }
  __shared__ double ls[256];
  __shared__ double lq[256];
  ls[threadIdx.x] = s;
  lq[threadIdx.x] = sq;
  __syncthreads();
  for (int off = 128; off > 0; off >>= 1) {
    if (threadIdx.x < off) {
      ls[threadIdx.x] += ls[threadIdx.x + off];
      lq[threadIdx.x] += lq[threadIdx.x + off];
    }
    __syncthreads();
  }
  if (threadIdx.x == 0) {
    part[(c * 16 + n) * 2 + 0] = ls[0];
    part[(c * 16 + n) * 2 + 1] = lq[0];
  }
}

// BN stage 2: fold 16 partials per channel (fixed order), emit mean and
// scale = gamma * rsqrt(var + eps) as float pairs.
__global__ void bn_finalize_kernel(const double* __restrict__ part,
                                   const float* __restrict__ gamma,
                                   float* __restrict__ stats)
{
  const int c = threadIdx.x;
  if (c < OC) {
    double s = 0.0, sq = 0.0;
    for (int n = 0; n < 16; ++n) {
      s  += part[(c * 16 + n) * 2 + 0];
      sq += part[(c * 16 + n) * 2 + 1];
    }
    const double cnt = (double)XN * (double)Y_CSTRIDE;  // 1,968,624
    const double mean = s / cnt;
    const double var  = sq / cnt - mean * mean;
    const double scale = (double)gamma[c] / sqrt(var + 1e-5);
    stats[c * 2 + 0] = (float)mean;
    stats[c * 2 + 1] = (float)scale;
  }
}

// BN stage 3: streamed in-place normalize of y.
__global__ __launch_bounds__(256) void bn_apply_kernel(
    float* __restrict__ y, const float* __restrict__ stats)
{
  const int i = blockIdx.x * 256 + threadIdx.x;
  if (i >= Y_CSTRIDE) return;
  const int c = blockIdx.y;
  const int n = blockIdx.z;
  const float mean  = stats[c * 2 + 0];
  const float scale = stats[c * 2 + 1];
  const size_t idx = ((size_t)n * OC + c) * Y_CSTRIDE + i;
  y[idx] = (y[idx] - mean) * scale;
}

extern "C" void kernel_launch(void* const* d_in, const int* in_sizes, int n_in,
                              void* d_out, int out_size, void* d_ws, size_t ws_size,
                              hipStream_t stream) {
  const float* x     = (const float*)d_in[0];
  const float* w     = (const float*)d_in[1];
  const float* gamma = (const float*)d_in[2];
  float* y = (float*)d_out;

  double* part = (double*)d_ws;                        // 32*16*2 doubles = 8KB
  float*  stats = (float*)((char*)d_ws + 8192);        // 32*2 floats

  // 4 D/H parity classes (both W parities fused); 8 waves/block, 1 tile/wave.
#define LAUNCH_CLASS(PD_, PH_)                                                 \
  {                                                                            \
    constexpr int Dq_ = (PD_) ? 15 : 16;                                       \
    constexpr int Hq_ = (PH_) ? 31 : 32;                                       \
    constexpr int tiles_ = XN * Dq_ * Hq_ * 2;                                 \
    convt_wmma_kernel<PD_, PH_>                                                \
        <<<(tiles_ + 7) / 8, 256, 0, stream>>>(x, w, y);                       \
  }
  LAUNCH_CLASS(0, 0)
  LAUNCH_CLASS(0, 1)
  LAUNCH_CLASS(1, 0)
  LAUNCH_CLASS(1, 1)
#undef LAUNCH_CLASS

  dim3 sg(OC, XN, 1);
  bn_stats_kernel<<<sg, 256, 0, stream>>>(y, part);
  bn_finalize_kernel<<<1, 64, 0, stream>>>(part, gamma, stats);
  dim3 ag((Y_CSTRIDE + 255) / 256, OC, XN);
  bn_apply_kernel<<<ag, 256, 0, stream>>>(y, stats);
}